// PCEncoding_81990925681065
// MI455X (gfx1250) — compile-verified
//
#include <hip/hip_runtime.h>
#include <hip/hip_bf16.h>

#define D 1024
#define STEP 5
#define BATCH 16
#define BS 80            // BATCH * STEP
#define ALPHA 0.01f
#define BN_EPS 1e-5f
#define L2_EPS 1e-12f

typedef __attribute__((ext_vector_type(2))) float v2f;
typedef __attribute__((ext_vector_type(8))) float v8f;

// ---------------------------------------------------------------------------
// Kernel 1: closed-form Hebbian readout.
// q[n] = alpha * sum_{s<=t} k_s * (k_s . x[n]) * prod_{r=s+1..t}(1 - alpha*k_r^2)
// where t = n/16 (step-major mem index), b = n%16, k_s = x[b*STEP + s].
// One block per output row n.
// ---------------------------------------------------------------------------
__global__ void __launch_bounds__(256)
compute_q_kernel(const float* __restrict__ x, float* __restrict__ q) {
    const int n   = blockIdx.x;      // 0..79
    const int t   = n / BATCH;       // step index 0..4
    const int bb  = n % BATCH;       // batch index 0..15
    const int tid = threadIdx.x;

    __shared__ float sdot[STEP];
    __shared__ float red[256];

    const float* xn = x + n * D;

    // dot products d_s = k_s . x[n], s = 0..t
    for (int s = 0; s <= t; ++s) {
        const float* ks = x + (bb * STEP + s) * D;
        float acc = 0.f;
        for (int i = tid; i < D; i += 256)
            acc = fmaf(ks[i], xn[i], acc);
        red[tid] = acc;
        __syncthreads();
        for (int off = 128; off > 0; off >>= 1) {
            if (tid < off) red[tid] += red[tid + off];
            __syncthreads();
        }
        if (tid == 0) sdot[s] = red[0];
        __syncthreads();
    }

    // elementwise recombination with decay products
    for (int i = tid; i < D; i += 256) {
        float acc   = 0.f;
        float decay = 1.f;                        // empty product at s = t
        for (int s = t; s >= 0; --s) {
            float ksi = x[(bb * STEP + s) * D + i];
            acc   = fmaf(ksi * sdot[s], decay, acc);
            decay *= (1.f - ALPHA * ksi * ksi);   // gain factor r = s
        }
        q[n * D + i] = ALPHA * acc;
    }
}

// ---------------------------------------------------------------------------
// Kernel 2: y = q @ W^T + bias via V_WMMA_F32_16X16X4_F32 (f32 in, f32 acc).
// Grid (8, 5): blockIdx.y = row tile (M), 8 waves/block each own a column
// tile (N).  A tile (16 x 128 K-chunk) staged in LDS with coalesced fill;
// B fragment read directly: B[k,n] = W[n,k], each lane streams one W row.
// ISA layouts: A 16x4 f32 -> lane M = lane&15, VGPR{0,1} = K {0,1}/{2,3} by
// lane half.  C/D: VGPR v holds M = v + 8*(lane>=16), N = lane&15.
// ---------------------------------------------------------------------------
__global__ void __launch_bounds__(256)
gemm_wmma_kernel(const float* __restrict__ q, const float* __restrict__ W,
                 const float* __restrict__ bias, float* __restrict__ y) {
    __shared__ float sA[16 * 128];

    const int tid  = threadIdx.x;
    const int lane = tid & 31;
    const int wave = tid >> 5;                 // 0..7
    const int tm   = blockIdx.y;               // 0..4
    const int tn   = blockIdx.x * 8 + wave;    // 0..63
    const int row0 = tm * 16;
    const int col0 = tn * 16;

    const int mA = lane & 15;                  // A row (== B col, same lane pattern)
    const int hi = lane >> 4;                  // 0 -> K{0,1}, 1 -> K{2,3}

    v8f acc = {};

    for (int kc = 0; kc < D; kc += 128) {
        // coalesced LDS fill of 16x128 A chunk (2048 elems, 8 per thread)
        for (int e = tid; e < 16 * 128; e += 256) {
            int r = e >> 7;
            int c = e & 127;
            sA[e] = q[(row0 + r) * D + kc + c];
        }
        __syncthreads();

        const float* wrow = W + (col0 + mA) * D + kc;   // B[k, mA] = W[mA-th col row]
        const float* arow = sA + mA * 128;

        #pragma unroll 8
        for (int k4 = 0; k4 < 128; k4 += 4) {
            v2f a, b;
            a.x = arow[k4 + 2 * hi + 0];
            a.y = arow[k4 + 2 * hi + 1];
            b.x = wrow[k4 + 2 * hi + 0];
            b.y = wrow[k4 + 2 * hi + 1];
            acc = __builtin_amdgcn_wmma_f32_16x16x4_f32(
                      /*neg_a=*/false, a, /*neg_b=*/false, b,
                      /*c_mod=*/(short)0, acc,
                      /*reuse_a=*/false, /*reuse_b=*/false);
        }
        __syncthreads();
    }

    // write D fragment + fused bias
    const int col = col0 + (lane & 15);
    const int rb  = row0 + (hi ? 8 : 0);
    const float bv = bias[col];
    #pragma unroll
    for (int v = 0; v < 8; ++v)
        y[(rb + v) * D + col] = acc[v] + bv;
}

// ---------------------------------------------------------------------------
// Kernel 3: per-column BatchNorm statistics (biased variance over 80 rows).
// One thread per column -> fully coalesced row sweeps.
// ---------------------------------------------------------------------------
__global__ void __launch_bounds__(256)
bn_stats_kernel(const float* __restrict__ y, float* __restrict__ mu,
                float* __restrict__ rstd) {
    const int c = blockIdx.x * 256 + threadIdx.x;   // 0..1023
    float s = 0.f, ss = 0.f;
    for (int r = 0; r < BS; ++r) {
        float v = y[r * D + c];
        s  += v;
        ss = fmaf(v, v, ss);
    }
    const float m   = s * (1.f / BS);
    const float var = ss * (1.f / BS) - m * m;
    mu[c]   = m;
    rstd[c] = rsqrtf(var + BN_EPS);
}

// ---------------------------------------------------------------------------
// Kernel 4: affine + ReLU + row-wise L2 normalize.  One block per row.
// ---------------------------------------------------------------------------
__global__ void __launch_bounds__(256)
finalize_kernel(const float* __restrict__ y, const float* __restrict__ mu,
                const float* __restrict__ rstd, const float* __restrict__ gamma,
                const float* __restrict__ beta, float* __restrict__ out) {
    const int n   = blockIdx.x;
    const int tid = threadIdx.x;
    __shared__ float red[256];

    float vals[4];
    float local = 0.f;
    #pragma unroll
    for (int j = 0; j < 4; ++j) {
        const int c = tid + j * 256;
        float v = (y[n * D + c] - mu[c]) * rstd[c] * gamma[c] + beta[c];
        v = fmaxf(v, 0.f);
        vals[j] = v;
        local = fmaf(v, v, local);
    }
    red[tid] = local;
    __syncthreads();
    for (int off = 128; off > 0; off >>= 1) {
        if (tid < off) red[tid] += red[tid + off];
        __syncthreads();
    }
    const float inv = 1.f / fmaxf(sqrtf(red[0]), L2_EPS);
    #pragma unroll
    for (int j = 0; j < 4; ++j)
        out[n * D + tid + j * 256] = vals[j] * inv;
}

// ---------------------------------------------------------------------------
extern "C" void kernel_launch(void* const* d_in, const int* in_sizes, int n_in,
                              void* d_out, int out_size, void* d_ws, size_t ws_size,
                              hipStream_t stream) {
    const float* x     = (const float*)d_in[0];   // [80, 1024]
    const float* W     = (const float*)d_in[1];   // [1024, 1024]
    const float* bias  = (const float*)d_in[2];   // [1024]
    const float* gamma = (const float*)d_in[3];   // [1024]
    const float* beta  = (const float*)d_in[4];   // [1024]
    float* out = (float*)d_out;                   // [80, 1024]

    // workspace layout (floats): q[80*1024] | y[80*1024] | mu[1024] | rstd[1024]
    float* q    = (float*)d_ws;
    float* y    = q + BS * D;
    float* mu   = y + BS * D;
    float* rstd = mu + D;

    compute_q_kernel<<<BS, 256, 0, stream>>>(x, q);

    dim3 gg(8, 5);   // 8 col-tile groups x 5 row tiles; 8 waves/block
    gemm_wmma_kernel<<<gg, 256, 0, stream>>>(q, W, bias, y);

    bn_stats_kernel<<<D / 256, 256, 0, stream>>>(y, mu, rstd);

    finalize_kernel<<<BS, 256, 0, stream>>>(y, mu, rstd, gamma, beta, out);
}